// Spatial_Module_25357486916383
// MI455X (gfx1250) — compile-verified
//
#include <hip/hip_runtime.h>
#include <math.h>

#define TS 4
#define BS 16
#define NN 2048
#define FF 64
#define NEGV (-9.0e15f)

typedef __attribute__((ext_vector_type(2))) float v2f;
typedef __attribute__((ext_vector_type(8))) float v8f;

// ---- block reductions over 256 threads via LDS tree ----
__device__ __forceinline__ float blk_sum(float v, float* red, int tid) {
    __syncthreads();                 // protect any prior read of red[0]
    red[tid] = v;
    __syncthreads();
    #pragma unroll
    for (int s = 128; s > 0; s >>= 1) {
        if (tid < s) red[tid] += red[tid + s];
        __syncthreads();
    }
    return red[0];
}

__device__ __forceinline__ float blk_max(float v, float* red, int tid) {
    __syncthreads();
    red[tid] = v;
    __syncthreads();
    #pragma unroll
    for (int s = 128; s > 0; s >>= 1) {
        if (tid < s) red[tid] = fmaxf(red[tid], red[tid + s]);
        __syncthreads();
    }
    return red[0];
}

// Kernel 1: for each (t, b) compute A[t,b] = sum_j softmax(att[2047, j]) * x[b, j]
// att[j] = v_j if (adj[2047,j] > 0 && v_j > 0) else NEG,  v_j = c1*x[b,2047] + c2*x[b,j]
__global__ void __launch_bounds__(256)
gat_row_reduce(const float* __restrict__ x,
               const float* __restrict__ adj,
               const float* __restrict__ W,
               const float* __restrict__ a,
               float* __restrict__ Aout) {
    const int blk = blockIdx.x;         // 0..63
    const int t   = blk >> 4;
    const int b   = blk & 15;
    const int tid = threadIdx.x;        // 0..255

    __shared__ float red[256];

    // c1 = dot(W_t, a1), c2 = dot(W_t, a2): threads 0..63 contribute one product each
    float p1 = 0.f, p2 = 0.f;
    if (tid < FF) {
        const float w = W[t * FF + tid];
        p1 = w * a[t * 2 * FF + tid];
        p2 = w * a[t * 2 * FF + FF + tid];
    }
    const float c1 = blk_sum(p1, red, tid);
    const float c2 = blk_sum(p2, red, tid);

    const float xi   = x[t * (BS * NN) + b * NN + (NN - 1)];
    const float base = c1 * xi;

    // each thread owns 8 coalesced j's, kept in registers (x read exactly once)
    float xv[8], att[8];
    float m = NEGV;
    #pragma unroll
    for (int k = 0; k < 8; ++k) {
        const int   j  = tid + k * 256;
        const float xj = x[t * (BS * NN) + b * NN + j];
        const float av = adj[(NN - 1) * NN + j];
        const float v  = base + c2 * xj;
        const float e  = (av > 0.f && v > 0.f) ? v : NEGV;
        xv[k]  = xj;
        att[k] = e;
        m = fmaxf(m, e);
    }
    const float M = blk_max(m, red, tid);

    // stable softmax accumulation (all-NEG row degrades to uniform weights, matching jax)
    float sw = 0.f, swx = 0.f;
    #pragma unroll
    for (int k = 0; k < 8; ++k) {
        const float w = __expf(att[k] - M);
        sw  += w;
        swx += w * xv[k];
    }
    const float SW  = blk_sum(sw,  red, tid);
    const float SWX = blk_sum(swx, red, tid);

    if (tid == 0) Aout[t * BS + b] = SWX / SW;
}

// Kernel 2: out[b, t, f] = elu(A[t,b] * W[t,f]) via V_WMMA_F32_16X16X4_F32.
// One wave (32 threads), EXEC all ones, no divergence around the WMMAs.
// A-matrix (16x4): documented layout -> K=0 lives in VGPR0 of lanes 0-15; all other
// K slots set to 0. B-matrix (4x16): replicate W[n] into every K slot (n = lane%16);
// since A is zero for K>0 the product is A[m]*W[n] independent of B's K-row mapping.
__global__ void __launch_bounds__(32)
gat_outer_wmma(const float* __restrict__ Ain,
               const float* __restrict__ W,
               float* __restrict__ out) {
    const int lane = threadIdx.x;     // 0..31
    const int n    = lane & 15;
    const int mbase = (lane < 16) ? 0 : 8;

    #pragma unroll
    for (int t = 0; t < TS; ++t) {
        const float Aval = Ain[t * BS + n];
        v2f amat;
        amat.x = (lane < 16) ? Aval : 0.f;   // K=0 (lanes 0-15, VGPR0); K=2 half zeroed
        amat.y = 0.f;                        // K=1 / K=3 zeroed

        #pragma unroll
        for (int tile = 0; tile < 4; ++tile) {
            const float wv = W[t * FF + tile * 16 + n];
            v2f bmat;
            bmat.x = wv;                     // every K row holds W[n]
            bmat.y = wv;

            v8f c = {};
            v8f d = __builtin_amdgcn_wmma_f32_16x16x4_f32(
                false, amat, false, bmat, (short)0, c, false, false);

            // C/D layout: lanes 0-15 -> N=lane, M=r; lanes 16-31 -> N=lane-16, M=8+r
            #pragma unroll
            for (int r = 0; r < 8; ++r) {
                const int  bb = mbase + r;
                const float v = d[r];
                const float o = (v > 0.f) ? v : (__expf(v) - 1.f);   // elu
                out[bb * (TS * FF) + t * FF + tile * 16 + n] = o;
            }
        }
    }
}

extern "C" void kernel_launch(void* const* d_in, const int* in_sizes, int n_in,
                              void* d_out, int out_size, void* d_ws, size_t ws_size,
                              hipStream_t stream) {
    (void)in_sizes; (void)n_in; (void)out_size; (void)ws_size;
    const float* x   = (const float*)d_in[0];   // (TS, BS*N, 1)
    const float* adj = (const float*)d_in[1];   // (N, N)
    const float* W   = (const float*)d_in[2];   // (TS, 1, F)
    const float* a   = (const float*)d_in[3];   // (TS, 2F, 1)
    float* out = (float*)d_out;                 // (BS, TS, F) = 4096 floats
    float* Aws = (float*)d_ws;                  // 64 floats of scratch

    gat_row_reduce<<<dim3(TS * BS), dim3(256), 0, stream>>>(x, adj, W, a, Aws);
    gat_outer_wmma<<<dim3(1), dim3(32), 0, stream>>>(Aws, W, out);
}